// GATLayer_11665131176088
// MI455X (gfx1250) — compile-verified
//
#include <hip/hip_runtime.h>
#include <math.h>

// Problem constants (B=16, N=2048, D_IN=D_OUT=256)
#define BB 16
#define NN 2048
#define DD 256

// LDS strides (floats):
//  XS=260 (mod 64 == 4)  : row-buffer stride, A-frag pattern conflict-free
//  CS=276 (mod 64 == 20) : row-major chunk, both access patterns conflict-free
//  SS=2052 (mod 64 == 4) : score stripe
//  PS=544                : pair-interleaved chunk stride (floats per pair-row);
//                          544 mod 64 == 32 -> lane-halves use disjoint banks
#define XS 260
#define CS 276
#define SS 2052
#define PS 544

// LDS chunk buffer size in kernel B (bytes): max(128*CS, 64*PS) floats
#define CHUNKB ((size_t)128 * CS * 4)

typedef float v2f __attribute__((ext_vector_type(2)));
typedef float v8f __attribute__((ext_vector_type(8)));

__device__ __forceinline__ v8f wmma4(v2f a, v2f b, v8f c) {
  return __builtin_amdgcn_wmma_f32_16x16x4_f32(
      /*neg_a=*/false, a, /*neg_b=*/false, b,
      /*c_mod=*/(short)0, c, /*reuse_a=*/false, /*reuse_b=*/false);
}

// CDNA5 async global->LDS copy (ASYNCcnt-tracked; bypasses VGPR round trip).
// Low 32 bits of a generic __shared__ pointer are the LDS byte offset.
__device__ __forceinline__ void async_ld_b128(const float* lds_dst, const float* gsrc) {
  unsigned off = (unsigned)(size_t)lds_dst;
  asm volatile("global_load_async_to_lds_b128 %0, %1, off"
               :: "v"(off), "v"(gsrc) : "memory");
}
__device__ __forceinline__ void wait_async0() {
  asm volatile("s_wait_asynccnt 0" ::: "memory");
}

// ---------------------------------------------------------------------------
// Kernel A: H[32768][256] = X[32768][256] * W[256][256]  (fp32 WMMA 16x16x4)
// Grid: 512 WGs x 256 threads; each WG: 64 rows x 256 cols.
// LDS order: Wp (hot B-frags) at offset 0 so DS immediate offsets fold.
// ---------------------------------------------------------------------------
__global__ void gat_h_kernel(const float* __restrict__ X,
                             const float* __restrict__ W,
                             float* __restrict__ H) {
  extern __shared__ char smem[];
  float* Wp = (float*)smem;                          // [32 pairs][PS] (69632 B)
  float* Xl = (float*)(smem + (size_t)32 * PS * 4);  // [64][XS]

  const int tid  = threadIdx.x;
  const int wave = tid >> 5;
  const int lane = tid & 31;
  const int half = lane >> 4;
  const int lr   = lane & 15;
  const long R0  = (long)blockIdx.x * 64;

  // Async-stage 64x256 X tile (row-major)
  for (int it = 0; it < 16; ++it) {
    int idx = tid + it * 256;
    int row = idx >> 6;
    int c4  = (idx & 63) << 2;
    async_ld_b128(Xl + row * XS + c4, X + (R0 + row) * DD + c4);
  }

  const int rt  = wave >> 1;        // row tile 0..3
  const int ct0 = (wave & 1) * 8;   // col tile base 0 or 8
  v8f acc[8];
  #pragma unroll
  for (int t = 0; t < 8; ++t)
    acc[t] = (v8f){0.f, 0.f, 0.f, 0.f, 0.f, 0.f, 0.f, 0.f};

  wait_async0();

  for (int kc = 0; kc < DD; kc += 64) {
    __syncthreads();
    // Build pair-interleaved W chunk: rows kc..kc+63 -> 32 pair-rows
    for (int it = 0; it < 16; ++it) {
      int li = tid + it * 256;       // 0..4095
      int p  = li >> 7;              // pair 0..31
      int n0 = (li & 127) << 1;      // 0..254 step 2
      const float* g0 = W + (kc + 2 * p) * DD + n0;
      float2 x0 = *(const float2*)g0;
      float2 x1 = *(const float2*)(g0 + DD);
      float4 v = {x0.x, x1.x, x0.y, x1.y};
      *(float4*)(Wp + p * PS + 2 * n0) = v;
    }
    __syncthreads();

    for (int kk = 0; kk < 64; kk += 4) {
      v2f a = *(const v2f*)(Xl + (rt * 16 + lr) * XS + kc + kk + 2 * half);
      const float* bp = Wp + ((kk >> 1) + half) * PS + 2 * (ct0 * 16 + lr);
      v2f bt[8];
      #pragma unroll
      for (int t = 0; t < 8; ++t)
        bt[t] = *(const v2f*)(bp + t * 32);   // 32 floats = 128 B imm steps
      #pragma unroll
      for (int t = 0; t < 8; ++t)
        acc[t] = wmma4(a, bt[t], acc[t]);
    }
  }

  #pragma unroll
  for (int t = 0; t < 8; ++t) {
    long col = (ct0 + t) * 16 + lr;
    #pragma unroll
    for (int i = 0; i < 8; ++i) {
      long row = R0 + rt * 16 + i + 8 * half;
      H[row * DD + col] = acc[t][i];
    }
  }
}

// ---------------------------------------------------------------------------
// Kernel B: per (batch, 16-row stripe):
//   S = lrelu(Hr Hᵀ) .* adj ; P = softmax(S) ; out = P * nodes + bias
// Grid: 2048 WGs x 256 threads.
// LDS order: ch (hot chunk) at 0, then scores, then h-rows (~282.5 KB total).
// ---------------------------------------------------------------------------
__global__ void gat_attn_kernel(const float* __restrict__ nodes,
                                const float* __restrict__ adj,
                                const float* __restrict__ bias,
                                const float* __restrict__ H,
                                float* __restrict__ out) {
  extern __shared__ char smem[];
  float* ch = (float*)smem;                                  // chunk (141312 B)
  float* s  = (float*)(smem + CHUNKB);                       // [16][SS]
  float* hr = (float*)(smem + CHUNKB + (size_t)16 * SS * 4); // [16][XS]

  const int tid  = threadIdx.x;
  const int wave = tid >> 5;
  const int lane = tid & 31;
  const int half = lane >> 4;
  const int lr   = lane & 15;

  const int b  = blockIdx.x >> 7;
  const int rt = blockIdx.x & 127;
  const long r0 = (long)b * NN + rt * 16;
  const float* Hb = H     + (long)b * NN * DD;
  const float* Nb = nodes + (long)b * NN * DD;
  const float* Ab = adj   + ((long)b * NN + rt * 16) * NN;

  // Async-stage this stripe's 16 rows of H
  for (int it = 0; it < 4; ++it) {
    int idx = tid + it * 256;
    int row = idx >> 6;
    int c4  = (idx & 63) << 2;
    async_ld_b128(hr + row * XS + c4, Hb + (rt * 16 + row) * DD + c4);
  }
  wait_async0();
  __syncthreads();

  // Hoist all 64 A-fragments (K=256) for the stripe into registers:
  // reused by all 16 column chunks. Occupancy is LDS-limited, VGPRs are free.
  v2f Afr[64];
  #pragma unroll
  for (int j = 0; j < 64; ++j)
    Afr[j] = *(const v2f*)(hr + lr * XS + 4 * j + 2 * half);

  // ---- Phase 1: scores S = lrelu(Hr Hᵀ) .* adj ----
  for (int cb = 0; cb < NN; cb += 128) {
    __syncthreads();
    // Async-stage H[cb..cb+127][0..255] row-major (B-frag pairs contiguous in k)
    for (int it = 0; it < 32; ++it) {
      int idx = tid + it * 256;
      int row = idx >> 6;
      int c4  = (idx & 63) << 2;
      async_ld_b128(ch + row * CS + c4, Hb + (long)(cb + row) * DD + c4);
    }
    wait_async0();
    __syncthreads();

    v8f acc = (v8f){0.f, 0.f, 0.f, 0.f, 0.f, 0.f, 0.f, 0.f};
    const float* bbase = ch + (wave * 16 + lr) * CS + 2 * half;
    #pragma unroll
    for (int j = 0; j < 64; j += 4) {
      v2f b0 = *(const v2f*)(bbase + 4 * j);
      v2f b1 = *(const v2f*)(bbase + 4 * j + 4);
      v2f b2 = *(const v2f*)(bbase + 4 * j + 8);
      v2f b3 = *(const v2f*)(bbase + 4 * j + 12);
      acc = wmma4(Afr[j + 0], b0, acc);
      acc = wmma4(Afr[j + 1], b1, acc);
      acc = wmma4(Afr[j + 2], b2, acc);
      acc = wmma4(Afr[j + 3], b3, acc);
    }

    const int mloc = wave * 16;
    #pragma unroll
    for (int i = 0; i < 8; ++i) {
      int row = i + 8 * half;
      float v = acc[i];
      v = v > 0.f ? v : 0.2f * v;                        // leaky_relu(0.2)
      float am = Ab[(long)row * NN + cb + mloc + lr];    // coalesced adj stream
      s[row * SS + cb + mloc + lr] = v * am;
    }
  }
  __syncthreads();

  // ---- Phase 2: row softmax (16 lanes per row, wave32 shfl reductions) ----
  {
    const int row = 2 * wave + half;
    float mx = -3.402823466e38f;
    for (int j = lr; j < NN; j += 16) mx = fmaxf(mx, s[row * SS + j]);
    #pragma unroll
    for (int off = 1; off < 16; off <<= 1)
      mx = fmaxf(mx, __shfl_xor(mx, off, 16));
    float sum = 0.f;
    for (int j = lr; j < NN; j += 16) {
      float e = __expf(s[row * SS + j] - mx);
      s[row * SS + j] = e;
      sum += e;
    }
    #pragma unroll
    for (int off = 1; off < 16; off <<= 1)
      sum += __shfl_xor(sum, off, 16);
    float inv = 1.f / sum;
    for (int j = lr; j < NN; j += 16) s[row * SS + j] *= inv;
  }
  __syncthreads();

  // ---- Phase 3: out = P * nodes + bias ----
  // nodes chunk stored pair-interleaved: Np[p][d] = (nodes[2p][d], nodes[2p+1][d])
  v8f o0 = (v8f){0.f, 0.f, 0.f, 0.f, 0.f, 0.f, 0.f, 0.f};
  v8f o1 = (v8f){0.f, 0.f, 0.f, 0.f, 0.f, 0.f, 0.f, 0.f};
  const int d0 = wave * 16;
  const int d1 = (wave + 8) * 16;

  for (int cb = 0; cb < NN; cb += 128) {
    __syncthreads();
    for (int it = 0; it < 32; ++it) {
      int li = tid + it * 256;       // 0..8191
      int p  = li >> 7;              // pair 0..63
      int dd = (li & 127) << 1;      // 0..254 step 2
      const float* g0 = Nb + (long)(cb + 2 * p) * DD + dd;
      float2 x0 = *(const float2*)g0;
      float2 x1 = *(const float2*)(g0 + DD);
      float4 v = {x0.x, x1.x, x0.y, x1.y};
      *(float4*)(ch + p * PS + 2 * dd) = v;
    }
    __syncthreads();

    const float* abase = s + lr * SS + cb + 2 * half;
    const float* nb0 = ch + half * PS + 2 * (d0 + lr);
    const float* nb1 = ch + half * PS + 2 * (d1 + lr);
    #pragma unroll
    for (int j = 0; j < 32; j += 2) {     // mk = 4j
      v2f a0 = *(const v2f*)(abase + 4 * j);
      v2f a1 = *(const v2f*)(abase + 4 * j + 4);
      v2f b00 = *(const v2f*)(nb0 + (2 * j) * PS);
      v2f b01 = *(const v2f*)(nb1 + (2 * j) * PS);
      v2f b10 = *(const v2f*)(nb0 + (2 * j + 2) * PS);
      v2f b11 = *(const v2f*)(nb1 + (2 * j + 2) * PS);
      o0 = wmma4(a0, b00, o0);
      o1 = wmma4(a0, b01, o1);
      o0 = wmma4(a1, b10, o0);
      o1 = wmma4(a1, b11, o1);
    }
  }

  const float bs0 = bias[d0 + lr];
  const float bs1 = bias[d1 + lr];
  #pragma unroll
  for (int i = 0; i < 8; ++i) {
    long row = r0 + i + 8 * half;
    out[row * DD + d0 + lr] = o0[i] + bs0;
    out[row * DD + d1 + lr] = o1[i] + bs1;
  }
}

// ---------------------------------------------------------------------------
extern "C" void kernel_launch(void* const* d_in, const int* in_sizes, int n_in,
                              void* d_out, int out_size, void* d_ws, size_t ws_size,
                              hipStream_t stream) {
  const float* nodes = (const float*)d_in[0]; // [16][2048][256]
  const float* adj   = (const float*)d_in[1]; // [16][2048][2048]
  const float* W     = (const float*)d_in[2]; // [256][256]
  const float* bias  = (const float*)d_in[3]; // [256]
  float* out = (float*)d_out;                 // [16][2048][256]
  float* H   = (float*)d_ws;                  // 33.5 MB scratch

  const size_t smemA = ((size_t)32 * PS + (size_t)64 * XS) * 4;          // ~133 KB
  const size_t smemB = CHUNKB + ((size_t)16 * SS + (size_t)16 * XS) * 4; // ~282.5 KB

  hipFuncSetAttribute(reinterpret_cast<const void*>(gat_h_kernel),
                      hipFuncAttributeMaxDynamicSharedMemorySize, (int)smemA);
  hipFuncSetAttribute(reinterpret_cast<const void*>(gat_attn_kernel),
                      hipFuncAttributeMaxDynamicSharedMemorySize, (int)smemB);

  gat_h_kernel<<<dim3(512), dim3(256), smemA, stream>>>(nodes, W, H);
  gat_attn_kernel<<<dim3(BB * (NN / 16)), dim3(256), smemB, stream>>>(nodes, adj, bias, H, out);
}